// surrogate_gradient_adder_dense_13786845020440
// MI455X (gfx1250) — compile-verified
//
#include <hip/hip_runtime.h>
#include <stdint.h>

// ---- CDNA5 TDM descriptor vector types (amdgpu-toolchain / clang-23 6-arg builtin) ----
typedef unsigned int u32x4 __attribute__((ext_vector_type(4)));
typedef int          i32x8 __attribute__((ext_vector_type(8)));
typedef int          i32x4 __attribute__((ext_vector_type(4)));

#define NB      16          // bits per value
#define LROWS   8           // L dimension
#define BPB     32          // batch rows per block (one per lane, wave32)
#define ROW_DW  (LROWS*NB)  // 128 dwords of payload per batch row
#define ROW_PAD (ROW_DW+4)  // +4 dword TDM pad -> 16B-aligned rows, stride 132

#if __has_builtin(__builtin_amdgcn_s_wait_tensorcnt)
#define WAIT_TENSORCNT0() __builtin_amdgcn_s_wait_tensorcnt(0)
#else
#define WAIT_TENSORCNT0() asm volatile("s_wait_tensorcnt 0x0" ::: "memory")
#endif

__global__ __launch_bounds__(32)
void sg_adder_dense_kernel(const float* __restrict__ x,
                           float* __restrict__ out,    // [B,16]
                           float* __restrict__ carry)  // [B,16]
{
    __shared__ __align__(16) float tile[BPB * ROW_PAD]; // 16896 B
    const int tid = threadIdx.x;
    const int b0  = blockIdx.x * BPB;

    // ---------------- TDM: stage 32 rows x 128 floats (16KB) global -> LDS ----------------
    // D# group0: count=1 | lds_addr | global_addr[56:0] | type=2
    {
        const uint64_t gaddr = (uint64_t)(uintptr_t)(x + (size_t)b0 * ROW_DW);
        const uint32_t laddr = (uint32_t)(uintptr_t)tile;   // low 32 bits of generic = LDS offset

        u32x4 g0;
        g0.x = 1u;                                           // [1:0] count=1 (valid)
        g0.y = laddr;                                        // [63:32] lds_addr
        g0.z = (uint32_t)gaddr;                              // [95:64] global_addr lo
        g0.w = (uint32_t)((gaddr >> 32) & 0x01FFFFFFull)     // [120:96] global_addr hi
             | (2u << 30);                                   // [127:126] type=2 (image)

        // D# group1
        i32x8 g1;
        g1[0] = (int)((2u << 16)      // data_size = 4B
                    | (1u << 20)      // pad_enable
                    | (6u << 22)      // pad_interval: code 6 -> every 128 dwords
                    | (3u << 25));    // pad_amount:  code 3 -> 4 dwords
        g1[1] = (int)(((uint32_t)(BPB * ROW_DW) & 0xFFFFu) << 16); // tensor_dim0[15:0] @ bits 63:48
        g1[2] = (int)(1u << 16);                                   // tensor_dim0 hi=0, tensor_dim1=1
        g1[3] = (int)(((uint32_t)(BPB * ROW_DW) & 0xFFFFu) << 16); // tile_dim0 = 4096
        g1[4] = 1;                                                 // tile_dim1=1, tile_dim2=0
        g1[5] = BPB * ROW_DW;                                      // tensor_dim0_stride lo = 4096
        g1[6] = 0;
        g1[7] = 0;

        i32x4 gz4 = {0, 0, 0, 0};                                  // groups 2/3: <=2D tensor
        i32x8 gz8 = {0, 0, 0, 0, 0, 0, 0, 0};                      // extra group (6-arg form)
        __builtin_amdgcn_tensor_load_to_lds(g0, g1, gz4, gz4, gz8, 0);
        WAIT_TENSORCNT0();
        asm volatile("" ::: "memory");   // LDS now holds the tile; forbid load hoisting
    }

    // ---------------- per-thread: one batch row b = b0 + tid ----------------
    const float* row = tile + tid * ROW_PAD;
    const float4* row4 = reinterpret_cast<const float4*>(row);

    float    xint[LROWS];
    uint32_t Abits[LROWS];
    float    xsum = 0.0f;

#pragma unroll
    for (int l = 0; l < LROWS; ++l) {
        float acc = 0.0f;
        uint32_t a = 0;
#pragma unroll
        for (int q = 0; q < 4; ++q) {                 // 4 x float4 = 16 values
            float4 v = row4[l * 4 + q];
            const int i0 = q * 4;
            acc = fmaf(v.x, (float)(1u << (i0 + 0)), acc);
            acc = fmaf(v.y, (float)(1u << (i0 + 1)), acc);
            acc = fmaf(v.z, (float)(1u << (i0 + 2)), acc);
            acc = fmaf(v.w, (float)(1u << (i0 + 3)), acc);
            a |= (v.x >= 0.5f ? 1u : 0u) << (i0 + 0);
            a |= (v.y >= 0.5f ? 1u : 0u) << (i0 + 1);
            a |= (v.z >= 0.5f ? 1u : 0u) << (i0 + 2);
            a |= (v.w >= 0.5f ? 1u : 0u) << (i0 + 3);
        }
        xint[l]  = acc;
        Abits[l] = a;
        xsum    += acc;
    }

    // ripple-carry adder collapses to integer identities:
    //   s_i  = bit i of (A+B)
    //   c_i  = bit (i+1) of (A ^ B ^ (A+B))   (carry-out of position i)
    int scnt[NB];
    int ccnt[NB];
#pragma unroll
    for (int i = 0; i < NB; ++i) { scnt[i] = 0; ccnt[i] = 0; }

#pragma unroll
    for (int l = 0; l < LROWS; ++l) {
        const int      resid = (int)(xsum - xint[l]);   // trunc toward zero (values >= 0)
        const uint32_t Bq = (uint32_t)resid & 0xFFFFu;
        const uint32_t Aq = Abits[l];
        const uint32_t T  = Aq + Bq;
        const uint32_t K  = (Aq ^ Bq ^ T) >> 1;
#pragma unroll
        for (int i = 0; i < NB; ++i) {
            scnt[i] += (int)((T >> i) & 1u);
            ccnt[i] += (int)((K >> i) & 1u);
        }
    }

    // ---------------- write out[b][0:16], carry[b][0:16] as b128 stores ----------------
    const int b = b0 + tid;
    float4* o4 = reinterpret_cast<float4*>(out   + (size_t)b * NB);
    float4* c4 = reinterpret_cast<float4*>(carry + (size_t)b * NB);
#pragma unroll
    for (int q = 0; q < 4; ++q) {
        float4 ov, cv;
        ov.x = scnt[q * 4 + 0] * 0.125f;  cv.x = (float)ccnt[q * 4 + 0];
        ov.y = scnt[q * 4 + 1] * 0.125f;  cv.y = (float)ccnt[q * 4 + 1];
        ov.z = scnt[q * 4 + 2] * 0.125f;  cv.z = (float)ccnt[q * 4 + 2];
        ov.w = scnt[q * 4 + 3] * 0.125f;  cv.w = (float)ccnt[q * 4 + 3];
        o4[q] = ov;
        c4[q] = cv;
    }
}

extern "C" void kernel_launch(void* const* d_in, const int* in_sizes, int n_in,
                              void* d_out, int out_size, void* d_ws, size_t ws_size,
                              hipStream_t stream) {
    (void)n_in; (void)out_size; (void)d_ws; (void)ws_size;
    const float* x = (const float*)d_in[0];
    const int total = in_sizes[0];                  // B * 8 * 16
    const int nB    = total / (LROWS * NB);         // 8192
    float* out   = (float*)d_out;                   // first output, B*16 floats
    float* carry = out + (size_t)nB * NB;           // second output, concatenated

    const int blocks = nB / BPB;                    // 8192/32 = 256 workgroups, 1 wave each
    sg_adder_dense_kernel<<<blocks, BPB, 0, stream>>>(x, out, carry);
}